// DBASolver_83184926589453
// MI455X (gfx1250) — compile-verified
//
#include <hip/hip_runtime.h>
#include <hip/hip_bf16.h>
#include <math.h>

typedef float v2f __attribute__((ext_vector_type(2)));
typedef float v8f __attribute__((ext_vector_type(8)));

#define THREADS_PER_BLOCK 256
#define WAVES_PER_BLOCK   8
#define SLOT_F            8     // floats per staged 7-vector (padded)
#define VECS_PER_WAVE     96    // 32 lanes * 3 vectors (u0, u1, v)

__device__ __forceinline__ float safe_lam(float lm) {
  if (lm != lm) lm = 100.0f;          // NaN -> 100
  return fmaxf(lm, 0.001f);
}

// ---------------------------------------------------------------------------
// Pass 0: zero the per-batch partial accumulators in workspace
// ---------------------------------------------------------------------------
__global__ void dba_zero(float* __restrict__ p, int n) {
  int i = blockIdx.x * blockDim.x + threadIdx.x;
  if (i < n) p[i] = 0.0f;
}

// ---------------------------------------------------------------------------
// Pass 1: streaming reduction.  Accumulates, per batch b:
//   M1 = sum_{n,c} u u^T,  u = sqrt(conf)*[Jp(6), r]      (H_pp | g_p)
//   M2 = sum_{n}   v v^T,  v = sqrt(invHdd)*[H_pd(6), g_d] (Schur term | g corr)
// via V_WMMA_F32_16X16X4_F32 (4 outer products per instruction), then
// atomically adds rows 0..5 / cols 0..6 of (M1 - M2) into partial[b][42].
// ---------------------------------------------------------------------------
__global__ void __launch_bounds__(THREADS_PER_BLOCK)
dba_pass1(const float* __restrict__ r, const float* __restrict__ w,
          const float* __restrict__ Jp, const float* __restrict__ Jd,
          const float* __restrict__ lmbda, float* __restrict__ partial,
          int N, int nodesPerBlock)
{
  __shared__ float lds[WAVES_PER_BLOCK * VECS_PER_WAVE * SLOT_F + 16]; // +pad for m<16 over-reads
  const int tid  = threadIdx.x;
  const int lane = tid & 31;
  const int wave = tid >> 5;
  const int b    = blockIdx.y;
  const long long nodeBase = (long long)blockIdx.x * nodesPerBlock;
  const float lm = safe_lam(lmbda[b]);

  float* V = &lds[wave * VECS_PER_WAVE * SLOT_F];

  v8f acc1 = {};   // sum u u^T
  v8f acc2 = {};   // sum v v^T

  const int iters = nodesPerBlock / THREADS_PER_BLOCK;
  for (int it = 0; it < iters; ++it) {
    long long node = nodeBase + (long long)it * THREADS_PER_BLOCK + tid;
    // out-of-range nodes contribute zero vectors (keeps EXEC uniform for WMMA)
    float conf = 0.f, nl = 0.f, r0 = 0.f, r1 = 0.f, jd0 = 0.f, jd1 = 0.f;
    float4 a0 = {0,0,0,0}, a1 = {0,0,0,0}, a2 = {0,0,0,0};
    if (node < N) {
      long long g = (long long)b * N + node;
      float2 ww = *(const float2*)(w  + 2 * g);
      float2 rr = *(const float2*)(r  + 2 * g);
      float2 jj = *(const float2*)(Jd + 2 * g);
      conf = ww.x; nl = ww.y; r0 = rr.x; r1 = rr.y; jd0 = jj.x; jd1 = jj.y;
      const float4* jp4 = (const float4*)(Jp + 12 * g);   // 48B-aligned
      a0 = jp4[0]; a1 = jp4[1]; a2 = jp4[2];
    }
    float jp0[6] = {a0.x, a0.y, a0.z, a0.w, a1.x, a1.y};  // c = 0
    float jp1[6] = {a1.z, a1.w, a2.x, a2.y, a2.z, a2.w};  // c = 1

    float sc  = __builtin_sqrtf(fmaxf(conf, 0.0f));
    float gd  = conf * (jd0 * r0 + jd1 * r1);
    float hdd = conf * (jd0 * jd0 + jd1 * jd1);
    float inv = 1.0f / fmaxf(hdd + lm + nl + 1e-4f, 1e-4f);
    float si  = __builtin_sqrtf(inv);
    float hpd[6];
    #pragma unroll
    for (int p = 0; p < 6; ++p) hpd[p] = conf * (jp0[p] * jd0 + jp1[p] * jd1);

    // stage: u0 -> slot lane, u1 -> slot 32+lane, v -> slot 64+lane
    float* s0 = V + (lane)      * SLOT_F;
    float* s1 = V + (32 + lane) * SLOT_F;
    float* s2 = V + (64 + lane) * SLOT_F;
    #pragma unroll
    for (int p = 0; p < 6; ++p) { s0[p] = sc * jp0[p]; s1[p] = sc * jp1[p]; s2[p] = si * hpd[p]; }
    s0[6] = sc * r0; s1[6] = sc * r1; s2[6] = si * gd;
    s0[7] = 0.f;     s1[7] = 0.f;     s2[7] = 0.f;

    // wave-coherent: all lanes' ds_stores complete before cross-lane ds_loads
    asm volatile("s_wait_dscnt 0" ::: "memory");

    // WMMA 16x16x4 f32 operand layout (per ISA):
    //   A: lanes 0-15 hold M=0..15 with VGPR0=K0,VGPR1=K1; lanes 16-31: K2,K3
    //   B: VGPR0: lanes0-15=K0,N ; lanes16-31=K1 ; VGPR1: K2/K3
    const int m = lane & 15;
    const int k = lane >> 4;
    #pragma unroll
    for (int grp = 0; grp < 16; ++grp) {           // 64 u-vectors -> acc1
      int base = grp * 4;
      v2f A, Bv;
      A.x  = V[(base + 2 * k    ) * SLOT_F + m];
      A.y  = V[(base + 2 * k + 1) * SLOT_F + m];
      Bv.x = V[(base + k        ) * SLOT_F + m];
      Bv.y = V[(base + 2 + k    ) * SLOT_F + m];
      acc1 = __builtin_amdgcn_wmma_f32_16x16x4_f32(false, A, false, Bv,
                                                   (short)0, acc1, false, false);
    }
    #pragma unroll
    for (int grp = 0; grp < 8; ++grp) {            // 32 v-vectors -> acc2
      int base = 64 + grp * 4;
      v2f A, Bv;
      A.x  = V[(base + 2 * k    ) * SLOT_F + m];
      A.y  = V[(base + 2 * k + 1) * SLOT_F + m];
      Bv.x = V[(base + k        ) * SLOT_F + m];
      Bv.y = V[(base + 2 + k    ) * SLOT_F + m];
      acc2 = __builtin_amdgcn_wmma_f32_16x16x4_f32(false, A, false, Bv,
                                                   (short)0, acc2, false, false);
    }
  }

  // C/D layout: lanes 0-15, VGPR j -> (M=j, N=lane).  Need M=0..5, N=0..6.
  if (lane < 7) {
    float* pb = partial + b * 42;
    #pragma unroll
    for (int j = 0; j < 6; ++j)
      atomicAdd(&pb[j * 7 + lane], acc1[j] - acc2[j]);
  }
}

// ---------------------------------------------------------------------------
// Pass 2: per-batch damping + 6x6 solve (16 tiny independent solves)
// ---------------------------------------------------------------------------
__global__ void dba_solve(const float* __restrict__ lmbda, const int* __restrict__ iter_idx,
                          const float* __restrict__ partial, float* __restrict__ dp_raw,
                          float* __restrict__ out_pose, int B)
{
  int b = threadIdx.x;
  if (b >= B) return;
  float H[6][7];
  for (int p = 0; p < 6; ++p) {
    for (int q = 0; q < 6; ++q) H[p][q] = partial[b * 42 + p * 7 + q];
    H[p][6] = partial[b * 42 + p * 7 + 6];          // g_eff
  }
  float lm = safe_lam(lmbda[b]);
  for (int p = 0; p < 6; ++p) H[p][p] += lm;        // + safe_lmbda * I
  if (*iter_idx >= 2) {
    for (int p = 3; p < 6; ++p) H[p][6] = 0.f;
    for (int p = 3; p < 6; ++p)
      for (int q = 3; q < 6; ++q) H[p][q] += 1e8f;
  }
  for (int p = 0; p < 6; ++p) H[p][p] += 1.0f;      // + I
  for (int p = 0; p < 6; ++p) H[p][p] *= 1.01f;     // + 0.01*diag on the diagonal

  // Gaussian elimination with partial pivoting on [H|g]
  for (int col = 0; col < 6; ++col) {
    int piv = col; float best = fabsf(H[col][col]);
    for (int q = col + 1; q < 6; ++q) {
      float v = fabsf(H[q][col]);
      if (v > best) { best = v; piv = q; }
    }
    if (piv != col)
      for (int q = col; q < 7; ++q) { float t = H[col][q]; H[col][q] = H[piv][q]; H[piv][q] = t; }
    float invd = 1.0f / H[col][col];
    for (int rI = col + 1; rI < 6; ++rI) {
      float f = H[rI][col] * invd;
      for (int q = col; q < 7; ++q) H[rI][q] -= f * H[col][q];
    }
  }
  float x[6];
  for (int p = 5; p >= 0; --p) {
    float s = H[p][6];
    for (int q = p + 1; q < 6; ++q) s -= H[p][q] * x[q];
    x[p] = s / H[p][p];
  }
  bool anynan = false;
  for (int p = 0; p < 6; ++p) { dp_raw[b * 6 + p] = x[p]; if (x[p] != x[p]) anynan = true; }
  for (int p = 0; p < 6; ++p) {
    float xp = anynan ? 0.f : fminf(fmaxf(x[p], -2.f), 2.f);
    out_pose[b * 6 + p] = xp;
  }
}

// ---------------------------------------------------------------------------
// Pass 3: per-node depth update (recomputes node terms, uses RAW delta_pose)
// ---------------------------------------------------------------------------
__global__ void __launch_bounds__(THREADS_PER_BLOCK)
dba_depth(const float* __restrict__ r, const float* __restrict__ w,
          const float* __restrict__ Jp, const float* __restrict__ Jd,
          const float* __restrict__ lmbda, const float* __restrict__ dp_raw,
          float* __restrict__ out_depth, int N)
{
  int b = blockIdx.y;
  long long node = (long long)blockIdx.x * blockDim.x + threadIdx.x;
  if (node >= N) return;
  long long g = (long long)b * N + node;
  float2 ww = *(const float2*)(w  + 2 * g);
  float2 rr = *(const float2*)(r  + 2 * g);
  float2 jj = *(const float2*)(Jd + 2 * g);
  const float4* jp4 = (const float4*)(Jp + 12 * g);
  float4 a0 = jp4[0], a1 = jp4[1], a2 = jp4[2];
  float jp0[6] = {a0.x, a0.y, a0.z, a0.w, a1.x, a1.y};
  float jp1[6] = {a1.z, a1.w, a2.x, a2.y, a2.z, a2.w};

  float conf = ww.x;
  float lm   = safe_lam(lmbda[b]);
  float gd   = conf * (jj.x * rr.x + jj.y * rr.y);
  float hdd  = conf * (jj.x * jj.x + jj.y * jj.y);
  float inv  = 1.0f / fmaxf(hdd + lm + ww.y + 1e-4f, 1e-4f);
  float dot  = 0.f;
  #pragma unroll
  for (int p = 0; p < 6; ++p) {
    float hpd = conf * (jp0[p] * jj.x + jp1[p] * jj.y);
    dot += hpd * dp_raw[b * 6 + p];
  }
  float dd = inv * (gd - dot);
  dd = (dd != dd) ? 0.f : fminf(fmaxf(dd, -5.f), 5.f);
  out_depth[g] = dd;
}

// ---------------------------------------------------------------------------
extern "C" void kernel_launch(void* const* d_in, const int* in_sizes, int n_in,
                              void* d_out, int out_size, void* d_ws, size_t ws_size,
                              hipStream_t stream) {
  const float* r     = (const float*)d_in[0];
  const float* w     = (const float*)d_in[1];
  const float* Jp    = (const float*)d_in[2];
  const float* Jd    = (const float*)d_in[3];
  const float* lmbda = (const float*)d_in[4];
  const int*   iter  = (const int*)d_in[5];

  const int B = 16, C = 2;
  const int N = in_sizes[0] / (B * C);          // 131072 for reference shapes

  float* ws      = (float*)d_ws;
  float* partial = ws;                          // 16 * 42 floats
  float* dp_raw  = ws + B * 42;                 // 16 * 6 floats
  float* out       = (float*)d_out;
  float* out_pose  = out;                       // (B, 6)
  float* out_depth = out + B * 6;               // (B, N, 1)

  dba_zero<<<dim3((B * 42 + 255) / 256), dim3(256), 0, stream>>>(partial, B * 42);

  int chunks = 32;
  int nodesPerBlock = ((N + chunks - 1) / chunks + THREADS_PER_BLOCK - 1)
                      / THREADS_PER_BLOCK * THREADS_PER_BLOCK;
  chunks = (N + nodesPerBlock - 1) / nodesPerBlock;
  dba_pass1<<<dim3(chunks, B), dim3(THREADS_PER_BLOCK), 0, stream>>>(
      r, w, Jp, Jd, lmbda, partial, N, nodesPerBlock);

  dba_solve<<<dim3(1), dim3(32), 0, stream>>>(lmbda, iter, partial, dp_raw, out_pose, B);

  dba_depth<<<dim3((N + THREADS_PER_BLOCK - 1) / THREADS_PER_BLOCK, B),
              dim3(THREADS_PER_BLOCK), 0, stream>>>(
      r, w, Jp, Jd, lmbda, dp_raw, out_depth, N);
}